// GumbelSinkhornDecoder_15934328668936
// MI455X (gfx1250) — compile-verified
//
#include <hip/hip_runtime.h>
#include <stdint.h>

// Gumbel-Sinkhorn, B=32, N=1024, 20 iters.
// State-collapsed formulation: s = s0 - R[i] - C[j];
//   R[i] = LSE_j(s0[i,j]-C[j]);  C[j] = LSE_i(s0[i,j]-R[i]).
// s0 (128 MB) stays L2-resident; only R/C (256 KB) are updated per sweep.

#define GS_B 32
#define GS_N 1024

// ---------- CDNA5 feature guards ----------
#if defined(__has_builtin)
#if __has_builtin(__builtin_amdgcn_global_load_async_to_lds_b128) && \
    __has_builtin(__builtin_amdgcn_s_wait_asynccnt)
#define GS_ASYNC 1
#endif
#if __has_builtin(__builtin_amdgcn_wmma_f32_16x16x4_f32)
#define GS_WMMA 1
#endif
#endif

// async-to-LDS builtin takes (int4 AS1*, int4 AS3*, imm offset, imm cpol)
typedef int gs_i4 __attribute__((vector_size(16)));
typedef gs_i4 __attribute__((address_space(1))) gs_gi4;  // global int4
typedef gs_i4 __attribute__((address_space(3))) gs_li4;  // LDS int4
typedef __attribute__((ext_vector_type(2))) float v2f;
typedef __attribute__((ext_vector_type(8))) float v8f;

// ---------- kernel 0: s0 = scores + gumbel (float4 elementwise) ----------
__global__ __launch_bounds__(256) void gs_add(const float* __restrict__ a,
                                              const float* __restrict__ g,
                                              float* __restrict__ o) {
  size_t i = (size_t)blockIdx.x * 256 + threadIdx.x;  // float4 index
  float4 x = ((const float4*)a)[i];
  float4 y = ((const float4*)g)[i];
  float4 r;
  r.x = x.x + y.x; r.y = x.y + y.y; r.z = x.z + y.z; r.w = x.w + y.w;
  ((float4*)o)[i] = r;
}

// ---------- kernel 1: row pass  R[row] = LSE_j(s0[row,j] - C[j]) ----------
// one wave32 per row; 32 elements per lane held in VGPRs; two-pass LSE.
__global__ __launch_bounds__(256) void gs_row(const float* __restrict__ s0,
                                              const float* __restrict__ C,
                                              float* __restrict__ R,
                                              int useC) {
  int wid  = (blockIdx.x << 3) + (threadIdx.x >> 5);   // global row 0..32767
  int lane = threadIdx.x & 31;
  const float* rp = s0 + (size_t)wid * GS_N;
  const float* cp = C + (size_t)(wid >> 10) * GS_N;    // per-batch C
  float v[32];
#pragma unroll
  for (int k = 0; k < 8; ++k) {
    int j = (k << 7) + (lane << 2);                    // 128*k + 4*lane
    float4 x = *(const float4*)(rp + j);
    if (useC) {
      float4 c4 = *(const float4*)(cp + j);
      x.x -= c4.x; x.y -= c4.y; x.z -= c4.z; x.w -= c4.w;
    }
    v[4 * k + 0] = x.x; v[4 * k + 1] = x.y;
    v[4 * k + 2] = x.z; v[4 * k + 3] = x.w;
  }
  float m = v[0];
#pragma unroll
  for (int i = 1; i < 32; ++i) m = fmaxf(m, v[i]);
#pragma unroll
  for (int off = 16; off > 0; off >>= 1) m = fmaxf(m, __shfl_xor(m, off, 32));
  float s = 0.f;
#pragma unroll
  for (int i = 0; i < 32; ++i) s += __expf(v[i] - m);
#pragma unroll
  for (int off = 16; off > 0; off >>= 1) s += __shfl_xor(s, off, 32);
  if (lane == 0) R[wid] = m + __logf(s);
}

// ---------- kernel 2: col pass  C[j] = LSE_i(s0[i,j] - R[i]) ----------
// block = 1024 thr = 32 waves; block owns (batch b, 32 columns).
// wave w owns rows [32w,32w+32); lane l owns column j0+l.
// Tile staged Global->LDS via CDNA5 async DMA (ASYNCcnt), then per-lane LSE,
// then 32-way cross-wave LSE combine in LDS.
__global__ __launch_bounds__(1024) void gs_col(const float* __restrict__ s0,
                                               const float* __restrict__ R,
                                               float* __restrict__ C) {
  extern __shared__ float tile[];        // 32 waves * 32*32 floats = 128 KB
  __shared__ float pm[32][32];
  __shared__ float pe[32][32];
  int w = threadIdx.x >> 5, l = threadIdx.x & 31;
  int b  = blockIdx.x >> 5;
  int j0 = (blockIdx.x & 31) << 5;
  const float* Sb = s0 + ((size_t)b << 20);
  float* myTile = tile + (w << 10);
  float v[32];
#if defined(GS_ASYNC)
  {
    int rsub = l >> 3;             // which of 4 rows in a 512B chunk
    int csub = (l & 7) << 2;       // float offset within row
#pragma unroll
    for (int q = 0; q < 8; ++q) {
      int r = (w << 5) + (q << 2) + rsub;
      const float* g = Sb + (size_t)r * GS_N + j0 + csub;
      float* dp = myTile + (((q << 2) + rsub) << 5) + csub;
      __builtin_amdgcn_global_load_async_to_lds_b128(
          (gs_gi4*)(unsigned long long)(uintptr_t)g,
          (gs_li4*)(unsigned)(uintptr_t)dp, 0, 0);
    }
    __builtin_amdgcn_s_wait_asynccnt(0);
#pragma unroll
    for (int r = 0; r < 32; ++r) v[r] = myTile[(r << 5) + l];
  }
#else
#pragma unroll
  for (int r = 0; r < 32; ++r)
    v[r] = Sb[(size_t)((w << 5) + r) * GS_N + j0 + l];
#endif
  // subtract R[row]; one coalesced load + broadcast via shuffle
  float Rl = R[(b << 10) + (w << 5) + l];
#pragma unroll
  for (int r = 0; r < 32; ++r) v[r] -= __shfl(Rl, r, 32);
  float m = v[0];
#pragma unroll
  for (int r = 1; r < 32; ++r) m = fmaxf(m, v[r]);
  float e = 0.f;
#pragma unroll
  for (int r = 0; r < 32; ++r) e += __expf(v[r] - m);
  pm[w][l] = m;
  pe[w][l] = e;
  __syncthreads();
  if (w == 0) {
    float M = pm[0][l];
#pragma unroll
    for (int q = 1; q < 32; ++q) M = fmaxf(M, pm[q][l]);
    float S = 0.f;
#pragma unroll
    for (int q = 0; q < 32; ++q) S += pe[q][l] * __expf(pm[q][l] - M);
    C[(b << 10) + j0 + l] = M + __logf(S);
  }
}

// ---------- kernel 3: out = exp(s0 - R[i] - C[j]) ----------
// One wave32 per 16x16 tile. Rank-1 broadcast (-R[i]-C[j]) done with
// v_wmma_f32_16x16x4_f32: A = [-R | 1 | 0 | 0], B = [1 ; -C ; 0 ; 0],
// C-acc = s0 tile (exact math; matches 16x16 f32 C/D VGPR layout).
__global__ __launch_bounds__(256) void gs_final(float* __restrict__ s0,
                                                const float* __restrict__ R,
                                                const float* __restrict__ C) {
  int wtile = (blockIdx.x << 3) + (threadIdx.x >> 5);  // 0..131071
  int lane  = threadIdx.x & 31;
  int b = wtile >> 12;          // 4096 tiles per batch (64x64)
  int t = wtile & 4095;
  int i0 = (t >> 6) << 4;
  int j0 = (t & 63) << 4;
  int half = lane >> 4;
  int ln   = lane & 15;
  float* Sb = s0 + ((size_t)b << 20);
  float Rv  = R[(b << 10) + i0 + ln];
  float Cvl = C[(b << 10) + j0 + ln];
  v8f c;
#pragma unroll
  for (int v = 0; v < 8; ++v)
    c[v] = Sb[(size_t)(i0 + v + (half << 3)) * GS_N + j0 + ln];
  v8f d;
#if defined(GS_WMMA)
  {
    v2f a, bb;
    a.x  = half ? 0.f : -Rv;   a.y  = half ? 0.f : 1.f;
    bb.x = half ? 0.f : 1.f;   bb.y = half ? 0.f : -Cvl;
    d = __builtin_amdgcn_wmma_f32_16x16x4_f32(false, a, false, bb,
                                              (short)0, c, false, false);
  }
#else
#pragma unroll
  for (int v = 0; v < 8; ++v)
    d[v] = c[v] - R[(b << 10) + i0 + v + (half << 3)] - Cvl;
#endif
#pragma unroll
  for (int v = 0; v < 8; ++v)
    Sb[(size_t)(i0 + v + (half << 3)) * GS_N + j0 + ln] = __expf(d[v]);
}

extern "C" void kernel_launch(void* const* d_in, const int* in_sizes, int n_in,
                              void* d_out, int out_size, void* d_ws, size_t ws_size,
                              hipStream_t stream) {
  const float* scores = (const float*)d_in[0];
  const float* noise  = (const float*)d_in[1];
  float* s0 = (float*)d_out;               // d_out doubles as s0 storage
  float* R  = (float*)d_ws;                // 32*1024 floats
  float* C  = R + GS_B * GS_N;             // 32*1024 floats (256 KB total)

  // s0 = scores + noise  (8388608 float4s)
  gs_add<<<32768, 256, 0, stream>>>(scores, noise, s0);

  for (int it = 0; it < 20; ++it) {
    gs_row<<<4096, 256, 0, stream>>>(s0, C, R, it > 0 ? 1 : 0);
    gs_col<<<1024, 1024, 32 * 1024 * (int)sizeof(float), stream>>>(s0, R, C);
  }

  // in-place: d_out = exp(s0 - R - C)
  gs_final<<<16384, 256, 0, stream>>>(s0, R, C);
}